// SubPixelUpscaling_11132555231954
// MI455X (gfx1250) — compile-verified
//
#include <hip/hip_runtime.h>
#include <cstdint>

// SubPixelUpscaling (pixel shuffle), r=2:
//   in : (8, 256, 256, 256) fp32
//   out: (8, 512, 512,  64) fp32
//   out[n, 2h+r1, 2w+r2, c] = in[n, h, w, 4c + 2r1 + r2]
//
// Pure permutation: ~1.07 GB traffic, roofline ~46us @ 23.3 TB/s.
// Strategy: async-DMA input tile into LDS (B128 both sides, padded
// conflict-free layout), shuffle through LDS, nontemporal B128 stores.

typedef float f32x4 __attribute__((ext_vector_type(4)));

#define THREADS       256
#define PIX_PER_TILE  32              // input pixels per block (32 KB payload)
#define GROUP_STRIDE  20              // 16 data floats + 4 pad floats (keeps 16B align, kills bank conflicts)
#define PIX_STRIDE    (16 * GROUP_STRIDE)          // 320 floats per input pixel in LDS
#define LDS_FLOATS    (PIX_PER_TILE * PIX_STRIDE)  // 10240 floats = 40 KB

__global__ __launch_bounds__(THREADS)
void subpixel_upscale_kernel(const float* __restrict__ in, float* __restrict__ out)
{
    __shared__ float lds[LDS_FLOATS];

    const int  t      = threadIdx.x;
    const long tileP0 = (long)blockIdx.x * PIX_PER_TILE;   // first input pixel of tile
    const float* gbase = in + tileP0 * 256;                // 256 floats per input pixel

    // ---------------- Load phase: 2048 x 16B chunks, 8 per thread ----------------
    // chunk m covers input floats [4m, 4m+3] of the tile (globally contiguous).
    // LDS placement: pixel p -> base 320p; 16-float group j -> +20j; chunk k -> +4k.
#pragma unroll
    for (int i = 0; i < 8; ++i) {
        const int m  = i * THREADS + t;
        const int p  = m >> 6;          // 64 chunks per input pixel
        const int mm = m & 63;
        const int j  = mm >> 2;         // 16-float group within pixel
        const int k  = mm & 3;          // 16B chunk within group

        const int fidx = p * PIX_STRIDE + j * GROUP_STRIDE + k * 4;
        const unsigned            lds_addr = (unsigned)(uintptr_t)(&lds[fidx]);
        const unsigned long long  gaddr    = (unsigned long long)(uintptr_t)(gbase + (long)m * 4);

        // GV-mode async copy: 16 bytes from global to LDS (ASYNCcnt)
        asm volatile("global_load_async_to_lds_b128 %0, %1, off"
                     :: "v"(lds_addr), "v"(gaddr)
                     : "memory");
    }

    asm volatile("s_wait_asynccnt 0x0" ::: "memory");
    __syncthreads();

    // ---------------- Store phase: 2048 output float4s, 8 per thread ----------------
    // o -> (input pixel p, quadrant q = 2*r1 + r2, channel group j).
    // out channels 4j..4j+3 <- input channels 16j + 4k + q, k = 0..3.
#pragma unroll
    for (int i = 0; i < 8; ++i) {
        const int o   = i * THREADS + t;
        const int p   = o >> 6;
        const int rem = o & 63;
        const int q   = rem >> 4;       // 0..3 = (r1,r2)
        const int j   = rem & 15;       // output channel group (4 floats)
        const int r1  = q >> 1;
        const int r2  = q & 1;

        const long P = tileP0 + p;      // global input pixel index
        const int  n = (int)(P >> 16);  // 256*256 pixels per image
        const int  h = (int)((P >> 8) & 255);
        const int  w = (int)(P & 255);

        const int base = p * PIX_STRIDE + j * GROUP_STRIDE + q;   // conflict-free across the wave
        f32x4 v;
        v.x = lds[base];
        v.y = lds[base + 4];
        v.z = lds[base + 8];
        v.w = lds[base + 12];

        const long oidx = ((((long)n * 512) + 2 * h + r1) * 512 + (2 * w + r2)) * 64 + 4 * j;
        __builtin_nontemporal_store(v, (f32x4*)(out + oidx));
    }
}

extern "C" void kernel_launch(void* const* d_in, const int* in_sizes, int n_in,
                              void* d_out, int out_size, void* d_ws, size_t ws_size,
                              hipStream_t stream)
{
    const float* x = (const float*)d_in[0];
    float*       y = (float*)d_out;

    // Fixed reference shapes: 8 * 256 * 256 pixels, 256 input channels each.
    const long totalPixels = (long)in_sizes[0] / 256;             // 524288
    const int  numTiles    = (int)(totalPixels / PIX_PER_TILE);   // 16384

    hipLaunchKernelGGL(subpixel_upscale_kernel,
                       dim3(numTiles), dim3(THREADS), 0, stream, x, y);
}